// Discriminator_12300786335758
// MI455X (gfx1250) — compile-verified
//
#include <hip/hip_runtime.h>

// ---------------------------------------------------------------------------
// Problem constants (from the reference)
// ---------------------------------------------------------------------------
#define LPn 4096
#define LHn 4096
#define Dn  1024
#define Hn  4096

typedef float v2f __attribute__((ext_vector_type(2)));
typedef float v8f __attribute__((ext_vector_type(8)));

// Tiling: block = 8 waves = 16(M) x 512(N); K staged in chunks of 16.
#define BN   512
#define KC   16
#define SAP  20   // sA row stride (floats): 80B -> float4-aligned, banks 20*lp distinct
#define SBP  520  // sB row stride (floats), row-major [k][n]: banks (8k+n)%64
#define SBTP 20   // sBT row stride (floats), [n][k] layout for transposed B

// ---------------------------------------------------------------------------
// CDNA5 async global->LDS copy (ASYNCcnt-tracked, no VGPR round-trip).
// LDS dest offset = low 32 bits of the generic shared-memory pointer
// (ISA 10.2: flat->LDS mapping truncates to addr[31:0]).
// ---------------------------------------------------------------------------
__device__ __forceinline__ void async_g2l_b128(void* ldsp, const void* gp) {
  unsigned int  loff = (unsigned int)(unsigned long long)(uintptr_t)ldsp;
  unsigned long long ga = (unsigned long long)(uintptr_t)gp;
  asm volatile("global_load_async_to_lds_b128 %0, %1, off"
               :: "v"(loff), "v"(ga) : "memory");
}
__device__ __forceinline__ void wait_async0() {
  asm volatile("s_wait_asynccnt 0x0" ::: "memory");
}

// ---------------------------------------------------------------------------
// Embedding gather: out[r, :] = W_emb[idx[r], :]   (float4 per thread)
// ---------------------------------------------------------------------------
__global__ void gather_rows(const float* __restrict__ W_emb,
                            const int* __restrict__ idx,
                            float* __restrict__ out,
                            int nrows, int ncols) {
  int t = blockIdx.x * blockDim.x + threadIdx.x;
  int q = ncols >> 2;
  int total = nrows * q;
  if (t >= total) return;
  int row = t / q;
  int c4  = t - row * q;
  const float4* src = (const float4*)(W_emb + (size_t)idx[row] * ncols);
  float4*       dst = (float4*)(out + (size_t)row * ncols);
  dst[c4] = src[c4];
}

// ---------------------------------------------------------------------------
// Unified fp32 WMMA GEMM.
//   out[m,n] = scale_m * ( sum_k Aelem(m,k) * Belem(k,n) ) (+bias[n]) (relu)
//   Aelem(m,k) = EXPA ? exp(rawA) : rawA;  rawA = TRANSA ? A[k*lda+m] : A[m*lda+k]
//   Belem(k,n) = TRANSB ? B[n*ldb+k] : B[k*ldb+n]
//   scale_m    = ROWSCALE ? 1/rowsum[m] : 1
// Panels staged in LDS via GLOBAL_LOAD_ASYNC_TO_LDS_B128 (ASYNCcnt), except
// the exp-fused A panel which must round-trip through VALU.
// Inner loop = ds_load + v_wmma only.
// ---------------------------------------------------------------------------
template <bool TRANSA, bool TRANSB, bool EXPA, bool HAS_BIAS, bool RELU, bool ROWSCALE>
__global__ __launch_bounds__(256, 2)
void wmma_gemm_f32(const float* __restrict__ A, int lda,
                   const float* __restrict__ B, int ldb,
                   const float* __restrict__ bias,
                   const float* __restrict__ rowsum,
                   float* __restrict__ C, int ldc,
                   int M, int N, int K) {
  __shared__ float sA[16 * SAP];
  __shared__ float sB[TRANSB ? (BN * SBTP) : (KC * SBP)];

  const int tid  = threadIdx.x;
  const int wave = tid >> 5;
  const int lane = tid & 31;
  const int half = lane >> 4;       // 0: K pair {0,1}, 1: K pair {2,3}
  const int lp   = lane & 15;

  const int m0  = blockIdx.y * 16;
  const int nb0 = blockIdx.x * BN;       // block's N origin
  const int nw  = wave * 64;             // wave's N offset within block

  float4* sA4 = (float4*)sA;
  float4* sB4 = (float4*)sB;

  // A needs the VALU path when exp() is fused or a transpose-scatter is needed.
  constexpr bool A_ASYNC = !EXPA && !TRANSA;

  // ---- Per-thread staging base pointers (advanced incrementally) ----
  const int aRow = (tid & 63) >> 2;      // 0..15 (m_local if !TRANSA, k_local if TRANSA)
  const int aC4  = tid & 3;              // 0..3
  const float* aStage =
      TRANSA ? (A + (size_t)aRow * lda + m0 + aC4 * 4)     // row k, contiguous m
             : (A + (size_t)(m0 + aRow) * lda + aC4 * 4);  // row m, contiguous k

  const int bR0 = tid >> 7;              // 0..1   (!TRANSB)
  const int bC4 = tid & 127;             // 0..127 (!TRANSB)
  const float* bStage =
      TRANSB ? (B + (size_t)(nb0 + tid * 2) * ldb)         // 2 rows of B per thread
             : (B + (size_t)bR0 * ldb + nb0 + bC4 * 4);    // rows bR0+2i, coalesced n

  v8f acc[4];
#pragma unroll
  for (int t = 0; t < 4; ++t) acc[t] = (v8f){0,0,0,0,0,0,0,0};

  for (int kc = 0; kc < K; kc += KC) {
    // ---- VALU-path A staging load: issued before the barrier (in flight) ----
    float4 av;
    if (!A_ASYNC) {
      if (tid < 64) av = *(const float4*)aStage;
    }

    __syncthreads();   // previous chunk fully consumed; LDS free for overwrite

    // ---- Async staging: global -> LDS directly, no VGPR round-trip ----
    if (A_ASYNC) {
      if (tid < 64)
        async_g2l_b128(&sA4[aRow * (SAP / 4) + aC4], aStage);
    }
    if (TRANSB) {
#pragma unroll
      for (int r = 0; r < 2; ++r)
#pragma unroll
        for (int c = 0; c < 4; ++c)
          async_g2l_b128(&sB4[(tid * 2 + r) * (SBTP / 4) + c],
                         bStage + (size_t)r * ldb + c * 4);
    } else {
#pragma unroll
      for (int i = 0; i < 8; ++i)
        async_g2l_b128(&sB4[(bR0 + 2 * i) * (SBP / 4) + bC4],
                       bStage + (size_t)(2 * i) * ldb);
    }

    // ---- VALU-path A staging store (exp fused / transpose scatter) ----
    if (!A_ASYNC) {
      if (tid < 64) {
        if (EXPA) {
          av.x = __expf(av.x); av.y = __expf(av.y);
          av.z = __expf(av.z); av.w = __expf(av.w);
        }
        if (TRANSA) {
          sA[(aC4 * 4 + 0) * SAP + aRow] = av.x;
          sA[(aC4 * 4 + 1) * SAP + aRow] = av.y;
          sA[(aC4 * 4 + 2) * SAP + aRow] = av.z;
          sA[(aC4 * 4 + 3) * SAP + aRow] = av.w;
        } else {
          sA4[aRow * (SAP / 4) + aC4] = av;
        }
      }
    }

    wait_async0();     // drain ASYNCcnt (barrier only covers DScnt)
    __syncthreads();

    // ---- Compute from LDS only ----
#pragma unroll
    for (int ks = 0; ks < KC; ks += 4) {
      const int ka = ks + 2 * half;
      const v2f a = *(const v2f*)(sA + lp * SAP + ka);
#pragma unroll
      for (int t = 0; t < 4; ++t) {
        const int nl = nw + t * 16 + lp;
        v2f b;
        if (TRANSB) {
          b = *(const v2f*)(sB + nl * SBTP + ka);
        } else {
          b.x = sB[ka * SBP + nl];
          b.y = sB[(ka + 1) * SBP + nl];
        }
        acc[t] = __builtin_amdgcn_wmma_f32_16x16x4_f32(
            false, a, false, b, (short)0, acc[t], false, false);
      }
    }

    // ---- Advance staging pointers ----
    aStage += TRANSA ? (size_t)KC * lda : (size_t)KC;
    bStage += TRANSB ? (size_t)KC : (size_t)KC * ldb;
  }

#pragma unroll
  for (int t = 0; t < 4; ++t) {
    const int n = nb0 + nw + t * 16 + lp;
    const float bn = HAS_BIAS ? bias[n] : 0.0f;
#pragma unroll
    for (int r = 0; r < 8; ++r) {
      const int m = m0 + r + 8 * half;
      float v = acc[t][r] + bn;
      if (RELU) v = fmaxf(v, 0.0f);
      if (ROWSCALE) v *= (1.0f / rowsum[m]);
      C[(size_t)m * ldc + n] = v;
    }
  }
}

// ---------------------------------------------------------------------------
// Column sums, two-stage (deterministic, 256 blocks of parallelism)
// ---------------------------------------------------------------------------
#define CS_RCHUNK 256
__global__ void col_partial(const float* __restrict__ E, float* __restrict__ part,
                            int N) {
  int j  = blockIdx.x * blockDim.x + threadIdx.x;
  int r0 = blockIdx.y * CS_RCHUNK;
  float s = 0.0f;
  for (int i = 0; i < CS_RCHUNK; ++i) s += E[(size_t)(r0 + i) * N + j];
  part[(size_t)blockIdx.y * N + j] = s;
}

__global__ void col_finish(const float* __restrict__ part, float* __restrict__ ekj,
                           int nparts, int N) {
  int j = blockIdx.x * blockDim.x + threadIdx.x;
  float s = 0.0f;
  for (int p = 0; p < nparts; ++p) s += part[(size_t)p * N + j];
  ekj[j] = s;
}

// eik[i] = sum_j E[i,j]  (block per row, LDS reduction)
__global__ void row_sums(const float* __restrict__ E, float* __restrict__ eik,
                         int N) {
  __shared__ float red[256];
  const int i = blockIdx.x;
  float s = 0.0f;
  for (int j = threadIdx.x; j < N; j += blockDim.x)
    s += E[(size_t)i * N + j];
  red[threadIdx.x] = s;
  __syncthreads();
  for (int off = 128; off > 0; off >>= 1) {
    if (threadIdx.x < off) red[threadIdx.x] += red[threadIdx.x + off];
    __syncthreads();
  }
  if (threadIdx.x == 0) eik[i] = red[0];
}

// ---------------------------------------------------------------------------
// fh single-row MLP (tiny: plain VALU is right here)
// ---------------------------------------------------------------------------
__global__ void fh_layer1(const float* __restrict__ p_last,
                          const float* __restrict__ W1,
                          const float* __restrict__ b1,
                          float* __restrict__ y1row) {
  int h = blockIdx.x * blockDim.x + threadIdx.x;
  float s = b1[h];
  for (int d = 0; d < Dn; ++d) s += p_last[d] * W1[(size_t)d * Hn + h];
  y1row[h] = fmaxf(s, 0.0f);
}

__global__ void fh_layer2(const float* __restrict__ y1row,
                          const float* __restrict__ W2,
                          const float* __restrict__ b2,
                          float* __restrict__ fhrow) {
  int h = blockIdx.x * blockDim.x + threadIdx.x;
  float s = b2[h];
  for (int k = 0; k < Hn; ++k) s += y1row[k] * W2[(size_t)k * Hn + h];
  fhrow[h] = fmaxf(s, 0.0f);
}

__global__ void bcast_rows(const float* __restrict__ row,
                           float* __restrict__ out,
                           int nrows, int ncols) {
  size_t t = (size_t)blockIdx.x * blockDim.x + threadIdx.x;
  int q = ncols >> 2;
  size_t total = (size_t)nrows * q;
  if (t >= total) return;
  int c4 = (int)(t % q);
  ((float4*)out)[t] = ((const float4*)row)[c4];
}

// ---------------------------------------------------------------------------
// Launch
// ---------------------------------------------------------------------------
extern "C" void kernel_launch(void* const* d_in, const int* in_sizes, int n_in,
                              void* d_out, int out_size, void* d_ws, size_t ws_size,
                              hipStream_t stream) {
  const int*   p_idx = (const int*)d_in[0];
  const int*   h_idx = (const int*)d_in[1];
  const float* W_emb = (const float*)d_in[2];
  const float* W1    = (const float*)d_in[3];
  const float* b1    = (const float*)d_in[4];
  const float* W2    = (const float*)d_in[5];
  const float* b2    = (const float*)d_in[6];

  float* out = (float*)d_out;
  // Output layout: fp[LP,H] | fh[LH,H] | E[LP,LH] | beta[LP,D] | alpha[LH,D]
  float* out_fp    = out;
  float* out_fh    = out + (size_t)LPn * Hn;
  float* out_E     = out_fh + (size_t)LHn * Hn;
  float* out_beta  = out_E + (size_t)LPn * LHn;
  float* out_alpha = out_beta + (size_t)LPn * Dn;

  // Workspace: p_emb | h_emb | eik | ekj | y1row | fhrow | cpart
  float* p_emb = (float*)d_ws;
  float* h_emb = p_emb + (size_t)LPn * Dn;
  float* eik   = h_emb + (size_t)LHn * Dn;
  float* ekj   = eik + LPn;
  float* y1row = ekj + LHn;
  float* fhrow = y1row + Hn;
  float* cpart = fhrow + Hn;             // (LPn/CS_RCHUNK) * LHn floats

  // 1) Gathers
  {
    int total = LPn * (Dn / 4);
    gather_rows<<<(total + 255) / 256, 256, 0, stream>>>(W_emb, p_idx, p_emb, LPn, Dn);
    gather_rows<<<(total + 255) / 256, 256, 0, stream>>>(W_emb, h_idx, h_emb, LHn, Dn);
  }

  // 2) y1 = relu(P @ W1 + b1) -> staged in the fh output region (overwritten later)
  {
    dim3 grid(Hn / BN, LPn / 16);
    wmma_gemm_f32<false, false, false, true, true, false><<<grid, 256, 0, stream>>>(
        p_emb, Dn, W1, Hn, b1, nullptr, out_fh, Hn, LPn, Hn, Dn);
  }

  // 3) fp = relu(y1 @ W2 + b2)
  {
    dim3 grid(Hn / BN, LPn / 16);
    wmma_gemm_f32<false, false, false, true, true, false><<<grid, 256, 0, stream>>>(
        out_fh, Hn, W2, Hn, b2, nullptr, out_fp, Hn, LPn, Hn, Hn);
  }

  // 4) fh row MLP (last premise embedding), broadcast overwrites the y1 staging
  fh_layer1<<<Hn / 256, 256, 0, stream>>>(p_emb + (size_t)(LPn - 1) * Dn, W1, b1, y1row);
  fh_layer2<<<Hn / 256, 256, 0, stream>>>(y1row, W2, b2, fhrow);
  {
    int total = LHn * (Hn / 4);
    bcast_rows<<<(total + 255) / 256, 256, 0, stream>>>(fhrow, out_fh, LHn, Hn);
  }

  // 5) E = P @ H^T  (transposed B; B panel async-staged [n][k] in LDS, no transpose)
  {
    dim3 grid(LHn / BN, LPn / 16);
    wmma_gemm_f32<false, true, false, false, false, false><<<grid, 256, 0, stream>>>(
        p_emb, Dn, h_emb, Dn, nullptr, nullptr, out_E, LHn, LPn, LHn, Dn);
  }

  // 6) Row / column sums of raw E (faithful to the reference's eik/ekj)
  row_sums<<<LPn, 256, 0, stream>>>(out_E, eik, LHn);
  {
    dim3 grid(LHn / 256, LPn / CS_RCHUNK);
    col_partial<<<grid, 256, 0, stream>>>(out_E, cpart, LHn);
    col_finish<<<LHn / 256, 256, 0, stream>>>(cpart, ekj, LPn / CS_RCHUNK, LHn);
  }

  // 7) beta = (exp(E)/eik) @ H   (exp fused into A staging)
  {
    dim3 grid(Dn / BN, LPn / 16);
    wmma_gemm_f32<false, false, true, false, false, true><<<grid, 256, 0, stream>>>(
        out_E, LHn, h_emb, Dn, nullptr, eik, out_beta, Dn, LPn, Dn, LHn);
  }

  // 8) alpha = (exp(E)^T/ekj) @ P  (A transpose-staged from E columns)
  {
    dim3 grid(Dn / BN, LHn / 16);
    wmma_gemm_f32<true, false, true, false, false, true><<<grid, 256, 0, stream>>>(
        out_E, LHn, p_emb, Dn, nullptr, ekj, out_alpha, Dn, LHn, Dn, LPn);
  }
}